// MPGNN_16492674417022
// MI455X (gfx1250) — compile-verified
//
#include <hip/hip_runtime.h>
#include <hip/hip_bf16.h>

// ---------------------------------------------------------------------------
// MPGNN (NNConv) for MI455X / gfx1250, wave32.
// All GEMMs use V_WMMA_F32_16X16X4_F32 (true fp32 matrix path).
// Edge-MLP is recomputed inside each message-passing step (fusion) to avoid
// materializing the 819MB per-edge weight tensor: ~200MB traffic per step
// instead of ~1.6GB, at ~15 GFLOP of WMMA per step (free at MI455X rates).
// ---------------------------------------------------------------------------

typedef float v2f __attribute__((ext_vector_type(2)));
typedef float v8f __attribute__((ext_vector_type(8)));

#define V_NODES   50000
#define E_EDGES   800000
#define NODE_IN   64
#define EDGE_IN   32
#define DM        16
#define EH        32
#define N_GRAPHS  128

static __device__ __forceinline__ v8f wmma4(v2f a, v2f b, v8f c) {
  // D = A(16x4 f32) * B(4x16 f32) + C(16x16 f32)
  return __builtin_amdgcn_wmma_f32_16x16x4_f32(
      /*neg_a=*/false, a, /*neg_b=*/false, b,
      /*c_mod=*/(short)0, c, /*reuse_a=*/false, /*reuse_b=*/false);
}

// ---------------------------------------------------------------------------
// Kernel 1: h = relu(node_feats @ W_proj + b_proj)   [V,64]@[64,16]
// One wave per 16-row tile; K=64 -> 16 WMMA k-steps. V/16 = 3125 tiles exactly.
// ---------------------------------------------------------------------------
__global__ void k_project(const float* __restrict__ nf,
                          const float* __restrict__ Wp,
                          const float* __restrict__ bp,
                          float* __restrict__ h) {
  const int lane = threadIdx.x & 31;
  const int wave = threadIdx.x >> 5;
  const int tile = blockIdx.x * (blockDim.x >> 5) + wave;
  const int half = lane >> 4;       // 0: lanes 0-15, 1: lanes 16-31
  const int n    = lane & 15;       // column / A-row selector

  const int arow = tile * 16 + n;   // A row (M = lane%16)
  v8f c = {0.f,0.f,0.f,0.f,0.f,0.f,0.f,0.f};

#pragma unroll
  for (int k0 = 0; k0 < NODE_IN; k0 += 4) {
    v2f a, b;
    // A 16x4: lanes0-15 K={k0,k0+1}, lanes16-31 K={k0+2,k0+3}
    a.x = nf[arow * NODE_IN + k0 + 2 * half];
    a.y = nf[arow * NODE_IN + k0 + 2 * half + 1];
    // B 4x16: lanes0-15 K low pair, lanes16-31 K high pair; col = lane%16
    b.x = Wp[(k0 + 2 * half)     * DM + n];
    b.y = Wp[(k0 + 2 * half + 1) * DM + n];
    c = wmma4(a, b, c);
  }

  const float bias = bp[n];
#pragma unroll
  for (int v = 0; v < 8; ++v) {
    const int m = tile * 16 + v + 8 * half;   // D row
    float x = c[v] + bias;
    h[m * DM + n] = x > 0.f ? x : 0.f;
  }
}

// ---------------------------------------------------------------------------
// Kernel 2 (the hot one): fused edge-MLP + per-edge matvec + scatter.
// One wave per 16-edge tile. E/16 = 50000 tiles exactly.
//   Z    = relu(ef_tile @ W_e1 + b_e1)          [16,32]   (WMMA, LDS restage)
//   Wetl = Z @ W_e2 + b_e2                      [16,256]  (WMMA, 16 N-blocks)
//   msg[e][j] = sum_i h[src[e]][i] * Wetl[e][i*16+j]
//   atomicAdd(agg[dst[e]][j], msg)
// N-block nb of the 256-wide output == input dim i; lane column == output j.
// ---------------------------------------------------------------------------
#define MP_WAVES 8
__global__ void k_mp_step(const float* __restrict__ ef,
                          const float* __restrict__ We1,
                          const float* __restrict__ be1,
                          const float* __restrict__ We2,
                          const float* __restrict__ be2,
                          const float* __restrict__ hin,
                          const int*   __restrict__ src,
                          const int*   __restrict__ dst,
                          float* agg) {
  __shared__ float lds[MP_WAVES][16 * EH + 16 * DM];   // Z tile + h[src] tile

  const int lane = threadIdx.x & 31;
  const int wave = threadIdx.x >> 5;
  const int tile = blockIdx.x * MP_WAVES + wave;
  const int e0   = tile * 16;
  const int half = lane >> 4;
  const int n    = lane & 15;

  float* zbuf = &lds[wave][0];          // [16][32] row-major
  float* hbuf = &lds[wave][16 * EH];    // [16][16] row-major

  // Stage h[src[e]] (16 edges x 16 feats) into LDS: 8 elements per lane.
#pragma unroll
  for (int t = 0; t < 8; ++t) {
    const int flat = t * 32 + lane;           // 0..255
    const int el   = flat >> 4;
    const int col  = flat & 15;
    hbuf[flat] = hin[src[e0 + el] * DM + col];
  }

  // ---- GEMM 1: Z = relu(ef @ W_e1 + b_e1), [16,32]@[32,32] ----------------
  v8f z0 = {0.f,0.f,0.f,0.f,0.f,0.f,0.f,0.f};
  v8f z1 = {0.f,0.f,0.f,0.f,0.f,0.f,0.f,0.f};
  const int arow = e0 + n;
  // prefetch next tile's edge features into cache (speculative)
  __builtin_prefetch(&ef[(arow + 16) * EDGE_IN], 0, 1);

#pragma unroll
  for (int k0 = 0; k0 < EDGE_IN; k0 += 4) {
    v2f a, b0, b1;
    a.x  = ef[arow * EDGE_IN + k0 + 2 * half];
    a.y  = ef[arow * EDGE_IN + k0 + 2 * half + 1];
    b0.x = We1[(k0 + 2 * half)     * EH + n];
    b0.y = We1[(k0 + 2 * half + 1) * EH + n];
    b1.x = We1[(k0 + 2 * half)     * EH + 16 + n];
    b1.y = We1[(k0 + 2 * half + 1) * EH + 16 + n];
    z0 = wmma4(a, b0, z0);
    z1 = wmma4(a, b1, z1);
  }
  const float bz0 = be1[n];
  const float bz1 = be1[16 + n];
#pragma unroll
  for (int v = 0; v < 8; ++v) {
    const int m = v + 8 * half;
    float x0 = z0[v] + bz0; x0 = x0 > 0.f ? x0 : 0.f;
    float x1 = z1[v] + bz1; x1 = x1 > 0.f ? x1 : 0.f;
    zbuf[m * EH + n]      = x0;
    zbuf[m * EH + 16 + n] = x1;
  }
  __syncthreads();   // uniform across the block; restage Z into A layout

  // ---- GEMM 2: Wetl = Z @ W_e2 + b_e2, [16,32]@[32,256] -------------------
  v8f acc[16];
#pragma unroll
  for (int nb = 0; nb < 16; ++nb)
    acc[nb] = (v8f){0.f,0.f,0.f,0.f,0.f,0.f,0.f,0.f};

#pragma unroll
  for (int k0 = 0; k0 < EH; k0 += 4) {
    v2f a;
    a.x = zbuf[n * EH + k0 + 2 * half];
    a.y = zbuf[n * EH + k0 + 2 * half + 1];
#pragma unroll
    for (int nb = 0; nb < 16; ++nb) {
      v2f b;
      b.x = We2[(k0 + 2 * half)     * (DM * DM) + nb * 16 + n];
      b.y = We2[(k0 + 2 * half + 1) * (DM * DM) + nb * 16 + n];
      acc[nb] = wmma4(a, b, acc[nb]);
    }
  }

  // ---- per-edge matvec + atomic scatter -----------------------------------
  // Lane handles output dim j = n for edges el = v + 8*half (v = 0..7).
#pragma unroll
  for (int v = 0; v < 8; ++v) {
    const int el = v + 8 * half;
    float m = 0.f;
#pragma unroll
    for (int nb = 0; nb < 16; ++nb) {
      m += hbuf[el * DM + nb] * (acc[nb][v] + be2[nb * 16 + n]);
    }
    atomicAdd(&agg[dst[e0 + el] * DM + n], m);
  }
}

// ---------------------------------------------------------------------------
// Kernel 3: h = relu(agg + conv_bias); re-zero agg for the next step.
// ---------------------------------------------------------------------------
__global__ void k_relu_bias(float* agg, const float* __restrict__ cbias,
                            float* __restrict__ hout, int nElem) {
  const int i = blockIdx.x * blockDim.x + threadIdx.x;
  if (i < nElem) {
    float x = agg[i] + cbias[i & 15];
    hout[i] = x > 0.f ? x : 0.f;
    agg[i]  = 0.f;
  }
}

// ---------------------------------------------------------------------------
// Kernel 4: per-graph mean pooling accumulation (sums + counts).
// Final relu(agg + conv_bias) fused in.
// ---------------------------------------------------------------------------
__global__ void k_pool(const float* __restrict__ agg,
                       const float* __restrict__ cbias,
                       const int*   __restrict__ gid,
                       float* sums, float* cnts, int nNodes) {
  const int i = blockIdx.x * blockDim.x + threadIdx.x;
  if (i < nNodes * DM) {
    const int v = i >> 4, d = i & 15;
    float x = agg[i] + cbias[d];
    x = x > 0.f ? x : 0.f;
    const int g = gid[v];
    atomicAdd(&sums[g * DM + d], x);
    if (d == 0) atomicAdd(&cnts[g], 1.0f);
  }
}

// ---------------------------------------------------------------------------
// Kernel 5: head: out[g][c] = mean(g) @ W_pred + b_pred.   [128,16]@[16,2]
// ---------------------------------------------------------------------------
__global__ void k_head(const float* __restrict__ sums,
                       const float* __restrict__ cnts,
                       const float* __restrict__ Wp,
                       const float* __restrict__ bp,
                       float* __restrict__ out, int nG) {
  const int i = blockIdx.x * blockDim.x + threadIdx.x;
  if (i < nG * 2) {
    const int g = i >> 1, c = i & 1;
    float cnt = cnts[g]; cnt = cnt > 1.f ? cnt : 1.f;
    const float inv = 1.f / cnt;
    float acc = bp[c];
#pragma unroll
    for (int d = 0; d < DM; ++d)
      acc += (sums[g * DM + d] * inv) * Wp[d * 2 + c];
    out[i] = acc;
  }
}

__global__ void k_zero(float* p, int n) {
  const int i = blockIdx.x * blockDim.x + threadIdx.x;
  if (i < n) p[i] = 0.f;
}

// ---------------------------------------------------------------------------
extern "C" void kernel_launch(void* const* d_in, const int* in_sizes, int n_in,
                              void* d_out, int out_size, void* d_ws, size_t ws_size,
                              hipStream_t stream) {
  const float* node_feats = (const float*)d_in[0];
  const float* edge_feats = (const float*)d_in[1];
  const float* W_proj     = (const float*)d_in[2];
  const float* b_proj     = (const float*)d_in[3];
  const float* W_e1       = (const float*)d_in[4];
  const float* b_e1       = (const float*)d_in[5];
  const float* W_e2       = (const float*)d_in[6];
  const float* b_e2       = (const float*)d_in[7];
  const float* conv_bias  = (const float*)d_in[8];
  const float* W_pred     = (const float*)d_in[9];
  const float* b_pred     = (const float*)d_in[10];
  const int*   src        = (const int*)d_in[11];
  const int*   dst        = (const int*)d_in[12];
  const int*   gid        = (const int*)d_in[13];

  float* ws  = (float*)d_ws;
  float* hA  = ws;                        // V*16 = 800000 floats
  float* hB  = ws + V_NODES * DM;         // 800000 floats
  float* agg = ws + 2 * V_NODES * DM;     // 800000 floats
  float* sums = agg + V_NODES * DM;       // 128*16 = 2048 floats
  float* cnts = sums + N_GRAPHS * DM;     // 128 floats

  const int nHE = V_NODES * DM;                       // 800000
  const int nZ  = nHE + N_GRAPHS * DM + N_GRAPHS;     // agg + sums + cnts (contiguous)

  // Zero accumulators (poisoned by harness; atomics require zero init each call)
  k_zero<<<(nZ + 255) / 256, 256, 0, stream>>>(agg, nZ);

  // h0 = relu(node_feats @ W_proj + b_proj)
  // 3125 tiles of 16 rows; 25 waves/block * 125 blocks = 3125 exactly
  k_project<<<125, 25 * 32, 0, stream>>>(node_feats, W_proj, b_proj, hA);

  // Step 1: fused edge-MLP + matvec + scatter. 50000 tiles / 8 waves = 6250 blocks.
  k_mp_step<<<6250, MP_WAVES * 32, 0, stream>>>(edge_feats, W_e1, b_e1, W_e2, b_e2,
                                                hA, src, dst, agg);
  // h1 = relu(agg + conv_bias); agg := 0
  k_relu_bias<<<(nHE + 255) / 256, 256, 0, stream>>>(agg, conv_bias, hB, nHE);

  // Step 2
  k_mp_step<<<6250, MP_WAVES * 32, 0, stream>>>(edge_feats, W_e1, b_e1, W_e2, b_e2,
                                                hB, src, dst, agg);

  // Pooling (fused relu+bias) and head
  k_pool<<<(nHE + 255) / 256, 256, 0, stream>>>(agg, conv_bias, gid, sums, cnts, V_NODES);
  k_head<<<1, 256, 0, stream>>>(sums, cnts, W_pred, b_pred, (float*)d_out, N_GRAPHS);
}